// NN_Decoder_5952824672800
// MI455X (gfx1250) — compile-verified
//
#include <hip/hip_runtime.h>
#include <hip/hip_bf16.h>

// Problem constants (from reference)
#define UNIT_NO 512
#define T_DIM   25
#define K_DIM   20
#define H_DIM   64
#define P_DIM   4096
#define F_DIM   64
#define BATCH   32
#define KTOT    (K_DIM * F_DIM)      // 1280 = hidden-layer reduction length
#define KC      128                  // W1 K-chunk rows staged per async transfer
#define NCHUNK  (KTOT / KC)          // 10
#define WBUF    (KC * H_DIM)         // 8192 floats per W1 LDS buffer
#define GOFF    (2 * WBUF)           // G starts after the two W1 buffers (64KB)
#define GFLOATS (BATCH * KTOT)       // 40960 floats of LDS for gathered G

typedef float v2f __attribute__((ext_vector_type(2)));
typedef float v8f __attribute__((ext_vector_type(8)));

// ---------------------------------------------------------------------------
// CDNA5 async global->LDS copy (ASYNCcnt-tracked DMA path), with fallback.
// Probe-verified: param 1 is `int4 addrspace(1)*`, param 2 `int4 addrspace(3)*`.
// ---------------------------------------------------------------------------
#if defined(__has_builtin)
#if __has_builtin(__builtin_amdgcn_global_load_async_to_lds_b128)
#define USE_ASYNC_LDS 1
#endif
#endif

#ifdef USE_ASYNC_LDS
typedef int v4i_ __attribute__((vector_size(16)));
typedef __attribute__((address_space(1))) v4i_* gv4p;   // global int4*
typedef __attribute__((address_space(3))) v4i_* lv4p;   // LDS int4*

__device__ __forceinline__ void async_copy16(const float* g, float* l) {
    // global_load_async_to_lds_b128: per-lane global addr + per-lane LDS addr
    __builtin_amdgcn_global_load_async_to_lds_b128((gv4p)g, (lv4p)l, 0, 0);
}
template <int N> __device__ __forceinline__ void wait_async() {
#if __has_builtin(__builtin_amdgcn_s_wait_asynccnt)
    __builtin_amdgcn_s_wait_asynccnt((unsigned short)N);
#else
    asm volatile("s_wait_asynccnt %0" :: "i"(N) : "memory");
#endif
}
#else
__device__ __forceinline__ void async_copy16(const float* g, float* l) {
    *(float4*)l = *(const float4*)g;          // synchronous fallback
}
template <int N> __device__ __forceinline__ void wait_async() {}
#endif

// ---------------------------------------------------------------------------
// Kernel 1: per-unit featurize  F[b,n,f] = sum_t S[b,n,t]*Wf[n,t,f] + bf[n,f]
// 52 MFLOP total -> plain VALU kernel; Wf (3.3MB) is L2-resident.
// ---------------------------------------------------------------------------
__global__ void featurize_kernel(const float* __restrict__ S,
                                 const float* __restrict__ Wf,
                                 const float* __restrict__ bf,
                                 float* __restrict__ F) {
    const int idx = blockIdx.x * blockDim.x + threadIdx.x;  // B*UNIT_NO*F_DIM
    const int f = idx & (F_DIM - 1);
    const int n = (idx >> 6) & (UNIT_NO - 1);
    const int b = idx >> 15;
    const float* s = S + (b * UNIT_NO + n) * T_DIM;
    const float* w = Wf + n * (T_DIM * F_DIM) + f;
    float acc = bf[n * F_DIM + f];
#pragma unroll
    for (int t = 0; t < T_DIM; ++t)
        acc = fmaf(s[t], w[t * F_DIM], acc);
    F[idx] = acc;
}

// ---------------------------------------------------------------------------
// Issue one W1 K-chunk (KC x 64 f32 = 32KB) as async DMA into LDS buffer c&1.
// 256 threads x 8 x b128 = 64 async wave-instructions per workgroup per chunk.
// ---------------------------------------------------------------------------
__device__ __forceinline__ void issue_w1_chunk(const float* __restrict__ W1p,
                                               float* lds, int c) {
    const int tid = threadIdx.x;
    const float* gsrc = W1p + (size_t)c * WBUF;
    float* ldst = lds + (c & 1) * WBUF;            // W1 buffers live at LDS 0
#pragma unroll
    for (int i = 0; i < WBUF / 4 / 256; ++i) {     // 8 float4 per thread
        const int j = (tid + i * 256) * 4;
        async_copy16(gsrc + j, ldst + j);
    }
}

// ---------------------------------------------------------------------------
// 32 WMMAs over one staged K-chunk: A from gathered G (LDS), B from W1 (LDS).
// With W1 buffers in the low 64KB of LDS, every B access offset fits the
// 16-bit DS immediate -> no per-iteration VALU address math.
// ---------------------------------------------------------------------------
__device__ __forceinline__ v8f compute_chunk(v8f acc, const float* lds, int c,
                                             int m, int n, int hi) {
    const float* Wb = lds + (c & 1) * WBUF + n;             // + r*64 folds to imm
    const float* Ga = lds + GOFF + m * KTOT + c * KC + 2 * hi;
#pragma unroll 4
    for (int kk = 0; kk < KC; kk += 4) {
        // A 16x4 f32: lanes 0-15 hold K=k,k+1; lanes 16-31 hold K=k+2,k+3
        const v2f av = *(const v2f*)(Ga + kk);
        // B 4x16 f32: rows striped across lanes; VGPR0 rows {0,2}, VGPR1 {1,3}
        const int r = kk + 2 * hi;
        v2f bv;
        bv.x = Wb[r * H_DIM];
        bv.y = Wb[(r + 1) * H_DIM];
        acc = __builtin_amdgcn_wmma_f32_16x16x4_f32(
                  false, av, false, bv, (short)0, acc, false, false);
    }
    return acc;
}

// ---------------------------------------------------------------------------
// Kernel 2: per-pixel decoder, one workgroup (8 wave32) per pixel.
//   async-gather G_p[32][1280] -> LDS; async double-buffered W1 stream -> LDS;
//   H = G_p x W1_p via v_wmma_f32_16x16x4_f32; +b1, PReLU; dot with Wo.
// LDS layout: [W1 buf0: 32KB][W1 buf1: 32KB][G: 160KB]  (224KB total)
// ---------------------------------------------------------------------------
__global__ void __launch_bounds__(256)
decoder_kernel(const float* __restrict__ F,          // [B, UNIT_NO, F_DIM]
               const int*   __restrict__ pix_units,  // [P, K_DIM]
               const float* __restrict__ W1,         // [P, KTOT, H_DIM]
               const float* __restrict__ b1,         // [P, H_DIM]
               const float* __restrict__ a,          // [P]
               const float* __restrict__ Wo,         // [P, H_DIM]
               const float* __restrict__ bo,         // [P]
               float*       __restrict__ out)        // [B, P]
{
    extern __shared__ float lds[];
    const int p   = blockIdx.x;
    const int tid = threadIdx.x;
    const float* W1p = W1 + (size_t)p * (KTOT * H_DIM);

    // ---- Prologue: async gather of G_p (per-lane DMA gather) ---------------
    // 20 units * 32 batch rows * 16 float4 = 10240 transfers, 40 per thread.
    for (int i = tid; i < K_DIM * BATCH * (F_DIM / 4); i += 256) {
        const int k   = i / (BATCH * (F_DIM / 4));
        const int rem = i - k * (BATCH * (F_DIM / 4));
        const int b   = rem >> 4;
        const int f4  = rem & 15;
        const int unit = pix_units[p * K_DIM + k];
        async_copy16(F + ((b * UNIT_NO + unit) * F_DIM) + f4 * 4,
                     lds + GOFF + b * KTOT + k * F_DIM + f4 * 4);
    }
    issue_w1_chunk(W1p, lds, 0);     // +8 async ops/lane
    issue_w1_chunk(W1p, lds, 1);     // +8 async ops/lane (peak ASYNCcnt = 56)
    wait_async<8>();                 // gather + chunk0 complete (in-order)
    __syncthreads();

    // ---- Main loop: WMMA on chunk c while DMA fills chunk c+1 --------------
    const int wave = tid >> 5;
    const int lane = tid & 31;
    const int mi   = wave >> 2;      // batch tile   0..1  (M = 32)
    const int ni   = wave & 3;       // hidden tile  0..3  (N = 64)
    const int lo   = lane & 15;
    const int hi   = lane >> 4;
    const int m = mi * 16 + lo;
    const int n = ni * 16 + lo;

    v8f acc = {};
    for (int c = 0; c < NCHUNK; ++c) {
        acc = compute_chunk(acc, lds, c, m, n, hi);
        __syncthreads();                            // all waves done with buf c&1
        if (c + 2 < NCHUNK) {
            issue_w1_chunk(W1p, lds, c + 2);        // refill freed buffer
            wait_async<8>();                        // chunk c+1 resident
            __syncthreads();
        } else if (c + 1 < NCHUNK) {
            wait_async<0>();                        // last chunk resident
            __syncthreads();
        }
    }

    // ---- Epilogue: bias + PReLU, stage H[32][64] in LDS (reuses G region) --
    const float slope = a[p];
    const float bias  = b1[p * H_DIM + n];
    float* Hs = lds + GOFF;
#pragma unroll
    for (int j = 0; j < 8; ++j) {
        // C/D layout: VGPR j -> M = j (lanes 0-15) / j+8 (lanes 16-31)
        const int mrow = mi * 16 + j + 8 * hi;
        float v = acc[j] + bias;
        v = (v >= 0.0f) ? v : slope * v;
        Hs[mrow * H_DIM + n] = v;
    }
    __syncthreads();

    // ---- Per-batch scalar output: I[b,p] = H[b,:].Wo[p,:] + bo[p] ----------
    if (tid < BATCH) {
        float r = bo[p];
        const float* wo = Wo + p * H_DIM;
#pragma unroll
        for (int h = 0; h < H_DIM; ++h)
            r = fmaf(Hs[tid * H_DIM + h], wo[h], r);
        out[tid * P_DIM + p] = r;
    }
}

// ---------------------------------------------------------------------------
extern "C" void kernel_launch(void* const* d_in, const int* in_sizes, int n_in,
                              void* d_out, int out_size, void* d_ws, size_t ws_size,
                              hipStream_t stream) {
    const float* S   = (const float*)d_in[0];
    const int*   pix = (const int*)  d_in[1];
    const float* Wf  = (const float*)d_in[2];
    const float* bf  = (const float*)d_in[3];
    const float* W1  = (const float*)d_in[4];
    const float* b1  = (const float*)d_in[5];
    const float* a   = (const float*)d_in[6];
    const float* Wo  = (const float*)d_in[7];
    const float* bo  = (const float*)d_in[8];
    float* out = (float*)d_out;
    float* F   = (float*)d_ws;   // [B, UNIT_NO, F_DIM] f32 = 4 MB scratch

    featurize_kernel<<<(BATCH * UNIT_NO * F_DIM) / 256, 256, 0, stream>>>(S, Wf, bf, F);

    const size_t shmem = (size_t)(GOFF + GFLOATS) * sizeof(float);  // 224 KB
    decoder_kernel<<<P_DIM, 256, shmem, stream>>>(F, pix, W1, b1, a, Wo, bo, out);
}